// Where2comm_9354438771161
// MI455X (gfx1250) — compile-verified
//
#include <hip/hip_runtime.h>
#include <hip/hip_bf16.h>
#include <stdint.h>

#define B_ 4
#define N_ 5
#define L_ 5
#define C_ 64
#define H_ 100
#define W_ 252
#define HW_ (H_ * W_)

#ifndef __has_builtin
#define __has_builtin(x) 0
#endif

#if defined(__HIP_DEVICE_COMPILE__) && defined(__gfx1250__) && \
    __has_builtin(__builtin_amdgcn_tensor_load_to_lds) &&      \
    __has_builtin(__builtin_amdgcn_s_wait_tensorcnt)
#define USE_TDM 1
#else
#define USE_TDM 0
#endif

typedef unsigned int u32x4 __attribute__((ext_vector_type(4)));
typedef int i32x4 __attribute__((ext_vector_type(4)));
typedef int i32x8 __attribute__((ext_vector_type(8)));

// ---------------------------------------------------------------------------
// Precompute: fold normalize_affine + grid-normalization into a pixel-space
// affine per (b, i, m):  ix = A0*w + A1*h + A2 ; iy = A3*w + A4*h + A5
// ---------------------------------------------------------------------------
__global__ void where2comm_coeff_kernel(const float* __restrict__ T,
                                        float* __restrict__ coef) {
    int idx = blockIdx.x * blockDim.x + threadIdx.x;
    if (idx >= B_ * N_ * N_) return;
    int m = idx % N_;
    int t = idx / N_;
    int i = t % N_;
    int b = t / N_;

    const float* Tm = T + (size_t)(((b * L_ + i) * L_ + m) * 16);
    float T00 = Tm[0], T01 = Tm[1], T03 = Tm[3];
    float T10 = Tm[4], T11 = Tm[5], T13 = Tm[7];

    const float Hh = (float)H_, Ww = (float)W_;
    float M00 = T00;
    float M01 = T01 * (Hh / Ww);
    float M02 = T03 * (2.0f / (0.8f * Ww));
    float M10 = T10 * (Ww / Hh);
    float M11 = T11;
    float M12 = T13 * (2.0f / (0.8f * Hh));

    float A0 = M00;
    float A1 = M01 * (Ww - 1.0f) / (Hh - 1.0f);
    float A2 = (M02 - M00 - M01 + 1.0f) * 0.5f * (Ww - 1.0f);
    float A3 = M10 * (Hh - 1.0f) / (Ww - 1.0f);
    float A4 = M11;
    float A5 = (M12 - M10 - M11 + 1.0f) * 0.5f * (Hh - 1.0f);

    float* o = coef + (size_t)idx * 8;
    o[0] = A0; o[1] = A1; o[2] = A2;
    o[3] = A3; o[4] = A4; o[5] = A5;
    o[6] = 0.0f; o[7] = 0.0f;
}

// ---------------------------------------------------------------------------
// Main kernel: one thread = one output pixel (h,w) for one (b,i); all 64
// channels accumulated in registers; max over the 5 source agents.
// Coefficients DMA'd into LDS via TENSOR_LOAD_TO_LDS; bilinear taps loaded
// as adjacent pairs so the backend merges them into b64 loads.
// ---------------------------------------------------------------------------
#define TW 32
#define THH 8

__global__ __launch_bounds__(256) void where2comm_warp_fuse_kernel(
    const float* __restrict__ xin, const float* __restrict__ coef,
    float* __restrict__ out) {
    __shared__ float lds_coef[N_ * 8];

    const int bz = blockIdx.z;  // = b*N + i
    const int b  = bz / N_;
    const float* cblk = coef + (size_t)bz * (N_ * 8);

#if USE_TDM
    if (threadIdx.x < 32) {
        uint64_t ga       = (uint64_t)(uintptr_t)cblk;
        uint32_t lds_addr = (uint32_t)(uintptr_t)(&lds_coef[0]);
        u32x4 g0;
        g0.x = 1u;                                               // count=1
        g0.y = lds_addr;                                         // lds_addr
        g0.z = (uint32_t)(ga & 0xFFFFFFFFu);                     // gaddr lo
        g0.w = (uint32_t)((ga >> 32) & 0x1FFFFFFu) | (2u << 30); // hi|type=2
        i32x8 g1;
        g1.s0 = (int)(2u << 16);        // data_size = 4B
        g1.s1 = (int)((N_ * 8) << 16);  // tensor_dim0[15:0]
        g1.s2 = (int)(1u << 16);        // tensor_dim1 = 1
        g1.s3 = (int)((N_ * 8) << 16);  // tile_dim0 = 40
        g1.s4 = 1;                      // tile_dim1 = 1
        g1.s5 = (int)(N_ * 8);          // tensor_dim0_stride
        g1.s6 = 0;
        g1.s7 = 0;
        i32x4 z4 = {0, 0, 0, 0};
        i32x8 z8 = {0, 0, 0, 0, 0, 0, 0, 0};
        __builtin_amdgcn_tensor_load_to_lds(g0, g1, z4, z4, z8, 0);
        __builtin_amdgcn_s_wait_tensorcnt(0);
    }
#else
    if (threadIdx.x < N_ * 8) lds_coef[threadIdx.x] = cblk[threadIdx.x];
#endif
    __syncthreads();

    const int w = blockIdx.x * TW + (int)(threadIdx.x % TW);
    const int h = blockIdx.y * THH + (int)(threadIdx.x / TW);
    if (w >= W_ || h >= H_) return;

    const float fw = (float)w, fh = (float)h;

    float acc[C_];
#pragma unroll
    for (int c = 0; c < C_; ++c) acc[c] = -3.402823466e38f;

    const float* xb = xin + (size_t)b * N_ * C_ * HW_;

    for (int m = 0; m < N_; ++m) {
        const float A0 = lds_coef[m * 8 + 0];
        const float A1 = lds_coef[m * 8 + 1];
        const float A2 = lds_coef[m * 8 + 2];
        const float A3 = lds_coef[m * 8 + 3];
        const float A4 = lds_coef[m * 8 + 4];
        const float A5 = lds_coef[m * 8 + 5];

        const float ix = fmaf(A1, fh, fmaf(A0, fw, A2));
        const float iy = fmaf(A4, fh, fmaf(A3, fw, A5));

        const float x0f = floorf(ix), y0f = floorf(iy);
        const int x0 = (int)x0f, y0 = (int)y0f;
        const float wx1 = ix - x0f, wy1 = iy - y0f;
        const float wx0 = 1.0f - wx1, wy0 = 1.0f - wy1;

        // validity folded into weights
        const float wx0v = (x0 >= 0 && x0 < W_) ? wx0 : 0.0f;
        const float wx1v = (x0 + 1 >= 0 && x0 + 1 < W_) ? wx1 : 0.0f;
        const float wy0v = (y0 >= 0 && y0 < H_) ? wy0 : 0.0f;
        const float wy1v = (y0 + 1 >= 0 && y0 + 1 < H_) ? wy1 : 0.0f;

        // Pair base in x: both taps loaded as adjacent floats [xb0, xb0+1].
        // Remap x-weights onto the loaded lanes (handles the +/-1 base shift
        // at the clamped edges; invalid taps already have zero weight).
        const int xb0 = min(max(x0, 0), W_ - 2);
        const float wAx = (xb0 == x0) ? wx0v : ((xb0 == x0 + 1) ? wx1v : 0.0f);
        const float wBx = (xb0 == x0) ? wx1v : ((xb0 + 1 == x0) ? wx0v : 0.0f);

        const int yc0 = min(max(y0, 0), H_ - 1);
        const int yc1 = min(max(y0 + 1, 0), H_ - 1);

        const float W00 = wAx * wy0v;  // row yc0, lane 0
        const float W01 = wBx * wy0v;  // row yc0, lane 1
        const float W10 = wAx * wy1v;  // row yc1, lane 0
        const float W11 = wBx * wy1v;  // row yc1, lane 1

        const int r0 = yc0 * W_ + xb0;
        const int r1 = yc1 * W_ + xb0;

        const float* src = xb + (size_t)m * C_ * HW_;
#pragma unroll
        for (int c = 0; c < C_; ++c) {
            const float* p0 = src + (size_t)c * HW_ + r0;
            const float* p1 = src + (size_t)c * HW_ + r1;
            const float a0 = p0[0], a1 = p0[1];  // provably adjacent -> b64
            const float b0 = p1[0], b1 = p1[1];
            float v = a0 * W00;
            v = fmaf(a1, W01, v);
            v = fmaf(b0, W10, v);
            v = fmaf(b1, W11, v);
            acc[c] = fmaxf(acc[c], v);
        }
    }

    float* o = out + (size_t)bz * C_ * HW_ + (size_t)(h * W_ + w);
#pragma unroll
    for (int c = 0; c < C_; ++c) o[(size_t)c * HW_] = acc[c];
}

// ---------------------------------------------------------------------------
extern "C" void kernel_launch(void* const* d_in, const int* in_sizes, int n_in,
                              void* d_out, int out_size, void* d_ws,
                              size_t ws_size, hipStream_t stream) {
    const float* x = (const float*)d_in[0];
    // d_in[1] (confidence_map) and d_in[2] (record_len) are unused by the ref.
    const float* T = (const float*)d_in[3];
    float* out = (float*)d_out;
    float* coef = (float*)d_ws;  // 100 * 8 floats = 3200 B

    where2comm_coeff_kernel<<<1, 128, 0, stream>>>(T, coef);

    dim3 grid((W_ + TW - 1) / TW, (H_ + THH - 1) / THH, B_ * N_);
    where2comm_warp_fuse_kernel<<<grid, 256, 0, stream>>>(x, coef, out);
}